// NormalGCN_15556371546754
// MI455X (gfx1250) — compile-verified
//
#include <hip/hip_runtime.h>
#include <hip/hip_bf16.h>

#define N_NODES 100000
#define N_EDGESC 1600000
#define DFEAT 128
#define NCLS 2

typedef float v2f __attribute__((ext_vector_type(2)));
typedef float v8f __attribute__((ext_vector_type(8)));

// ---------------- degree / normalization ----------------

__global__ void k_init_deg(float* __restrict__ deg) {
    int i = blockIdx.x * blockDim.x + threadIdx.x;
    if (i < N_NODES) deg[i] = 1.0f;   // self-loop contributes 1
}

__global__ void k_accum_deg(const int* __restrict__ dst, float* __restrict__ deg) {
    int e = blockIdx.x * blockDim.x + threadIdx.x;
    if (e < N_EDGESC) atomicAdd(&deg[dst[e]], 1.0f);
}

__global__ void k_make_dinv(float* __restrict__ deg) {
    int i = blockIdx.x * blockDim.x + threadIdx.x;
    if (i < N_NODES) deg[i] = rsqrtf(deg[i]);   // deg >= 1 always
}

// ---------------- h1 = x @ W1 via V_WMMA_F32_16X16X4_F32 ----------------
// Block = 256 threads = 8 waves. Block handles 16 rows; wave w handles cols [16w,16w+16).
// A (16x4 f32):  lanes 0-15 -> M=lane, VGPR0=K0,VGPR1=K1 ; lanes 16-31 -> K2,K3.
// B (4x16 f32):  VGPR0: lanes 0-15 = row K0 (N=lane), lanes 16-31 = row K2; VGPR1: K1/K3.
// C/D (16x16):   VGPR r: lanes 0-15 -> M=r, N=lane ; lanes 16-31 -> M=r+8.
__global__ __launch_bounds__(256) void k_gemm1_wmma(const float* __restrict__ x,
                                                    const float* __restrict__ W,
                                                    float* __restrict__ h) {
    const int wave  = threadIdx.x >> 5;
    const int lane  = threadIdx.x & 31;
    const int l15   = lane & 15;
    const int khalf = lane >> 4;           // 0 or 1
    const int row0  = blockIdx.x * 16;
    const int col0  = wave * 16;

    const float* xrow = x + (size_t)(row0 + l15) * DFEAT;
    v8f c = {};
    for (int k = 0; k < DFEAT; k += 4) {
        const int ka = k + khalf * 2;
        v2f a, b;
        a.x = xrow[ka];
        a.y = xrow[ka + 1];
        b.x = W[(size_t)(ka + 0) * DFEAT + col0 + l15];
        b.y = W[(size_t)(ka + 1) * DFEAT + col0 + l15];
        c = __builtin_amdgcn_wmma_f32_16x16x4_f32(
                /*neg_a=*/false, a, /*neg_b=*/false, b,
                /*c_mod=*/(short)0, c, /*reuse_a=*/false, /*reuse_b=*/false);
    }
    float* hout = h + (size_t)(row0 + khalf * 8) * DFEAT + col0 + l15;
#pragma unroll
    for (int r = 0; r < 8; ++r)
        hout[(size_t)r * DFEAT] = c[r];
}

// ---------------- aggregation layer 1 ----------------

__global__ void k_init_agg(float* __restrict__ agg, const float* __restrict__ b1) {
    int i = blockIdx.x * blockDim.x + threadIdx.x;
    if (i < N_NODES * DFEAT) agg[i] = b1[i & (DFEAT - 1)];
}

// one wave32 per (edge or self-loop); each lane moves a float4 (128 floats / 32 lanes)
__global__ __launch_bounds__(256) void k_scatter1(const int* __restrict__ src,
                                                  const int* __restrict__ dst,
                                                  const float* __restrict__ dinv,
                                                  const float* __restrict__ h1,
                                                  float* __restrict__ agg) {
    const int e    = (blockIdx.x * blockDim.x + threadIdx.x) >> 5;
    const int lane = threadIdx.x & 31;
    if (e >= N_EDGESC + N_NODES) return;
    int s, d;
    if (e < N_EDGESC) { s = src[e]; d = dst[e]; }
    else              { s = d = e - N_EDGESC; }
    const float w = dinv[s] * dinv[d];
    const float4 v = *(const float4*)(h1 + (size_t)s * DFEAT + lane * 4);
    float* out = agg + (size_t)d * DFEAT + lane * 4;
    atomicAdd(out + 0, v.x * w);
    atomicAdd(out + 1, v.y * w);
    atomicAdd(out + 2, v.z * w);
    atomicAdd(out + 3, v.w * w);
}

// ---------------- fused ReLU + (relu(agg) @ W2) : one wave per node ----------------

__global__ __launch_bounds__(256) void k_linear2(const float* __restrict__ agg,
                                                 const float* __restrict__ W2,
                                                 float* __restrict__ h2) {
    const int n    = (blockIdx.x * blockDim.x + threadIdx.x) >> 5;
    const int lane = threadIdx.x & 31;
    if (n >= N_NODES) return;
    const float4 v = *(const float4*)(agg + (size_t)n * DFEAT + lane * 4);
    const float a0 = fmaxf(v.x, 0.0f), a1 = fmaxf(v.y, 0.0f);
    const float a2 = fmaxf(v.z, 0.0f), a3 = fmaxf(v.w, 0.0f);
    const int j = lane * 4;
    float c0 = a0 * W2[(j + 0) * NCLS + 0] + a1 * W2[(j + 1) * NCLS + 0]
             + a2 * W2[(j + 2) * NCLS + 0] + a3 * W2[(j + 3) * NCLS + 0];
    float c1 = a0 * W2[(j + 0) * NCLS + 1] + a1 * W2[(j + 1) * NCLS + 1]
             + a2 * W2[(j + 2) * NCLS + 1] + a3 * W2[(j + 3) * NCLS + 1];
#pragma unroll
    for (int off = 16; off > 0; off >>= 1) {
        c0 += __shfl_down(c0, off);
        c1 += __shfl_down(c1, off);
    }
    if (lane == 0) {
        h2[(size_t)n * NCLS + 0] = c0;
        h2[(size_t)n * NCLS + 1] = c1;
    }
}

// ---------------- layer 2 aggregation into d_out ----------------

__global__ void k_init_out(float* __restrict__ out, const float* __restrict__ b2) {
    int i = blockIdx.x * blockDim.x + threadIdx.x;
    if (i < N_NODES * NCLS) out[i] = b2[i & (NCLS - 1)];
}

__global__ void k_scatter2(const int* __restrict__ src,
                           const int* __restrict__ dst,
                           const float* __restrict__ dinv,
                           const float* __restrict__ h2,
                           float* __restrict__ out) {
    int e = blockIdx.x * blockDim.x + threadIdx.x;
    if (e >= N_EDGESC + N_NODES) return;
    int s, d;
    if (e < N_EDGESC) { s = src[e]; d = dst[e]; }
    else              { s = d = e - N_EDGESC; }
    const float w = dinv[s] * dinv[d];
    atomicAdd(&out[(size_t)d * NCLS + 0], h2[(size_t)s * NCLS + 0] * w);
    atomicAdd(&out[(size_t)d * NCLS + 1], h2[(size_t)s * NCLS + 1] * w);
}

// ---------------- launcher ----------------

extern "C" void kernel_launch(void* const* d_in, const int* in_sizes, int n_in,
                              void* d_out, int out_size, void* d_ws, size_t ws_size,
                              hipStream_t stream) {
    const float* x    = (const float*)d_in[0];
    const int*   ei   = (const int*)d_in[1];      // [2, E] flat
    const float* W1   = (const float*)d_in[2];
    const float* b1   = (const float*)d_in[3];
    const float* W2   = (const float*)d_in[4];
    const float* b2   = (const float*)d_in[5];
    const int* src = ei;
    const int* dst = ei + N_EDGESC;

    float* ws   = (float*)d_ws;
    float* dinv = ws;                                   // N
    float* h1   = dinv + N_NODES;                       // N*128
    float* agg1 = h1 + (size_t)N_NODES * DFEAT;         // N*128
    float* h2   = agg1 + (size_t)N_NODES * DFEAT;       // N*2
    float* outp = (float*)d_out;

    const int B = 256;
    // normalization
    k_init_deg  <<<(N_NODES + B - 1) / B, B, 0, stream>>>(dinv);
    k_accum_deg <<<(N_EDGESC + B - 1) / B, B, 0, stream>>>(dst, dinv);
    k_make_dinv <<<(N_NODES + B - 1) / B, B, 0, stream>>>(dinv);
    // layer 1 linear (WMMA)
    k_gemm1_wmma<<<N_NODES / 16, B, 0, stream>>>(x, W1, h1);
    // layer 1 propagate
    k_init_agg  <<<((size_t)N_NODES * DFEAT + B - 1) / B, B, 0, stream>>>(agg1, b1);
    {
        long long waves = (long long)N_EDGESC + N_NODES;
        long long blocks = (waves * 32 + B - 1) / B;
        k_scatter1<<<(unsigned)blocks, B, 0, stream>>>(src, dst, dinv, h1, agg1);
    }
    // relu + layer 2 linear
    {
        long long blocks = ((long long)N_NODES * 32 + B - 1) / B;
        k_linear2<<<(unsigned)blocks, B, 0, stream>>>(agg1, W2, h2);
    }
    // layer 2 propagate into output
    k_init_out<<<(N_NODES * NCLS + B - 1) / B, B, 0, stream>>>(outp, b2);
    k_scatter2<<<((N_EDGESC + N_NODES) + B - 1) / B, B, 0, stream>>>(src, dst, dinv, h2, outp);
}